// SignGNN_11476152615592
// MI455X (gfx1250) — compile-verified
//
#include <hip/hip_runtime.h>

typedef __attribute__((ext_vector_type(2))) float v2f;
typedef __attribute__((ext_vector_type(8))) float v8f;

#define BN_EPS 1e-5f
#define NEG_SLOPE 0.01f

// ---------------------------------------------------------------- utilities

__global__ void fill_kernel(float* __restrict__ p, long long n, float v) {
    long long i = (long long)blockIdx.x * blockDim.x + threadIdx.x;
    if (i < n) p[i] = v;
}

// deg[dst] += 1 for every real edge (self-loops pre-seeded as deg=1)
__global__ void deg_kernel(const long long* __restrict__ ei, float* __restrict__ deg,
                           long long E) {
    long long e = (long long)blockIdx.x * blockDim.x + threadIdx.x;
    if (e < E) atomicAdd(&deg[ei[E + e]], 1.0f);
}

// deg -> d^{-1/2} in place (deg >= 1 always because of self loops)
__global__ void dinv_kernel(float* __restrict__ deg, int N) {
    int i = blockIdx.x * blockDim.x + threadIdx.x;
    if (i < N) deg[i] = rsqrtf(deg[i]);
}

__global__ void norm_kernel(const long long* __restrict__ ei, const float* __restrict__ dinv,
                            float* __restrict__ norm, long long E) {
    long long e = (long long)blockIdx.x * blockDim.x + threadIdx.x;
    if (e < E) norm[e] = dinv[ei[e]] * dinv[ei[E + e]];
}

// ---------------------------------------------------------------- layer 1 GEMM (K=3)

__global__ void gemm3_kernel(const float* __restrict__ x, const float* __restrict__ W,
                             float* __restrict__ out, int N) {
    long long idx = (long long)blockIdx.x * blockDim.x + threadIdx.x;
    if (idx >= (long long)N * 64) return;
    int n = (int)(idx >> 6);
    int f = (int)(idx & 63);
    const float* xr = x + (long long)n * 3;
    out[idx] = fmaf(xr[0], W[f], fmaf(xr[1], W[64 + f], xr[2] * W[128 + f]));
}

// ---------------------------------------------------------------- WMMA GEMM (fp32, K=FIN)
// One wave32 computes one 16x16 tile of Out = A[N,FIN] @ W[FIN,FOUT]
// using V_WMMA_F32_16X16X4_F32 chained over K.

template <int FIN, int FOUT>
__global__ __launch_bounds__(256) void gemm_wmma_kernel(const float* __restrict__ A,
                                                        const float* __restrict__ W,
                                                        float* __restrict__ Out, int N) {
    const int lane = threadIdx.x & 31;
    const int wave = threadIdx.x >> 5;
    const int colTiles = FOUT / 16;
    const int numRowTiles = (N + 15) >> 4;
    int tile = blockIdx.x * 8 + wave;
    if (tile >= numRowTiles * colTiles) return;  // uniform per wave -> EXEC stays full

    const int rowTile = tile / colTiles;
    const int colTile = tile - rowTile * colTiles;
    const int rowBase = rowTile << 4;
    const int colBase = colTile << 4;

    const int m = lane & 15;     // row (A) / col (B,D) within tile
    const int half = lane >> 4;  // which K-pair this lane holds

    int row = rowBase + m;
    if (row >= N) row = N - 1;  // clamp loads; stores predicated below
    const float* __restrict__ arow = A + (long long)row * FIN;

    v8f acc = {};
#pragma unroll
    for (int k = 0; k < FIN; k += 4) {
        const int ka = k + half * 2;
        v2f a, b;
        a.x = arow[ka + 0];
        a.y = arow[ka + 1];
        b.x = W[(ka + 0) * FOUT + colBase + m];
        b.y = W[(ka + 1) * FOUT + colBase + m];
        acc = __builtin_amdgcn_wmma_f32_16x16x4_f32(
            /*neg_a=*/false, a, /*neg_b=*/false, b,
            /*c_mod=*/(short)0, acc, /*reuse_a=*/false, /*reuse_b=*/false);
    }

#pragma unroll
    for (int v = 0; v < 8; ++v) {
        const int rr = rowBase + v + half * 8;  // 16x16 f32 C/D VGPR layout
        if (rr < N) Out[(long long)rr * FOUT + colBase + m] = acc[v];
    }
}

// ---------------------------------------------------------------- edge scatter
// accum[dst,f] += tmp[src,f] * norm[e]; one thread per (edge, feature).

template <int F, int LOGF>
__global__ __launch_bounds__(256) void scatter_kernel(const float* __restrict__ tmp,
                                                      const long long* __restrict__ ei,
                                                      const float* __restrict__ norm,
                                                      float* __restrict__ accum, long long E) {
    long long idx = (long long)blockIdx.x * blockDim.x + threadIdx.x;
    if (idx >= (E << LOGF)) return;
    long long e = idx >> LOGF;
    int f = (int)(idx & (F - 1));
    long long s = ei[e];
    long long d = ei[E + e];
    atomicAdd(&accum[d * F + f], tmp[s * F + f] * norm[e]);
}

// ---------------------------------------------------------------- epilogue
// out = lrelu(BN(accum + tmp*dinv^2 (self-loop) + bias))

template <int F, int LOGF>
__global__ __launch_bounds__(256) void epilogue_kernel(
    const float* __restrict__ tmp, const float* __restrict__ accum,
    const float* __restrict__ dinv, const float* __restrict__ bias,
    const float* __restrict__ gam, const float* __restrict__ bet,
    const float* __restrict__ rm, const float* __restrict__ rv,
    float* __restrict__ out, int N) {
    long long idx = (long long)blockIdx.x * blockDim.x + threadIdx.x;
    if (idx >= ((long long)N << LOGF)) return;
    int n = (int)(idx >> LOGF);
    int f = (int)(idx & (F - 1));
    float di = dinv[n];
    float y = accum[idx] + tmp[idx] * di * di + bias[f];
    float sc = gam[f] * rsqrtf(rv[f] + BN_EPS);
    y = (y - rm[f]) * sc + bet[f];
    out[idx] = (y >= 0.0f) ? y : NEG_SLOPE * y;
}

// ---------------------------------------------------------------- pooling

__global__ void pool_kernel(const float* __restrict__ h, const long long* __restrict__ batch,
                            float* __restrict__ pool, int N) {
    long long idx = (long long)blockIdx.x * blockDim.x + threadIdx.x;
    if (idx >= (long long)N * 32) return;
    int n = (int)(idx >> 5);
    int f = (int)(idx & 31);
    atomicAdd(&pool[batch[n] * 32 + f], h[idx]);
}

__global__ void cnt_kernel(const long long* __restrict__ batch, float* __restrict__ cnt, int N) {
    int n = blockIdx.x * blockDim.x + threadIdx.x;
    if (n < N) atomicAdd(&cnt[batch[n]], 1.0f);
}

__global__ void final_kernel(const float* __restrict__ pool, const float* __restrict__ cnt,
                             float* __restrict__ out, int G) {
    int idx = blockIdx.x * blockDim.x + threadIdx.x;
    if (idx >= G * 32) return;
    out[idx] = pool[idx] / fmaxf(cnt[idx >> 5], 1.0f);
}

// ---------------------------------------------------------------- launch

static inline unsigned nblk(long long n) { return (unsigned)((n + 255) / 256); }

extern "C" void kernel_launch(void* const* d_in, const int* in_sizes, int n_in,
                              void* d_out, int out_size, void* d_ws, size_t ws_size,
                              hipStream_t stream) {
    const float* x   = (const float*)d_in[0];
    const float* W1  = (const float*)d_in[1];
    const float* b1  = (const float*)d_in[2];
    const float* g1  = (const float*)d_in[3];
    const float* be1 = (const float*)d_in[4];
    const float* rm1 = (const float*)d_in[5];
    const float* rv1 = (const float*)d_in[6];
    const float* W2  = (const float*)d_in[7];
    const float* b2  = (const float*)d_in[8];
    const float* g2  = (const float*)d_in[9];
    const float* be2 = (const float*)d_in[10];
    const float* rm2 = (const float*)d_in[11];
    const float* rv2 = (const float*)d_in[12];
    const float* W3  = (const float*)d_in[13];
    const float* b3  = (const float*)d_in[14];
    const float* g3  = (const float*)d_in[15];
    const float* be3 = (const float*)d_in[16];
    const float* rm3 = (const float*)d_in[17];
    const float* rv3 = (const float*)d_in[18];
    const long long* ei    = (const long long*)d_in[19];
    const long long* batch = (const long long*)d_in[20];

    const int N = in_sizes[0] / 3;
    const long long E = (long long)in_sizes[19] / 2;
    const int G = out_size / 32;

    // workspace layout (floats)
    float* ws = (float*)d_ws;
    size_t off = 0;
    auto alloc = [&](size_t nel) {
        float* p = ws + off;
        off += (nel + 63) & ~(size_t)63;
        return p;
    };
    float* dinv = alloc((size_t)N);            // degree -> d^{-1/2}
    float* norm = alloc((size_t)E);            // per-edge coefficient
    float* hA   = alloc((size_t)N * 64);       // layer input / output
    float* hB   = alloc((size_t)N * 64);       // tmp = h @ W
    float* hC   = alloc((size_t)N * 64);       // scatter accumulator
    float* pool = alloc((size_t)G * 32);
    float* cnt  = alloc((size_t)G);
    (void)ws_size; (void)n_in;

    const dim3 B(256);

    // --- normalization coefficients
    fill_kernel<<<nblk(N), B, 0, stream>>>(dinv, N, 1.0f);  // self-loop degree
    deg_kernel<<<nblk(E), B, 0, stream>>>(ei, dinv, E);
    dinv_kernel<<<nblk(N), B, 0, stream>>>(dinv, N);
    norm_kernel<<<nblk(E), B, 0, stream>>>(ei, dinv, norm, E);

    // --- layer 1: x(3) @ W1 -> 64, scatter, BN+LReLU
    gemm3_kernel<<<nblk((long long)N * 64), B, 0, stream>>>(x, W1, hB, N);
    fill_kernel<<<nblk((long long)N * 64), B, 0, stream>>>(hC, (long long)N * 64, 0.0f);
    scatter_kernel<64, 6><<<nblk(E * 64), B, 0, stream>>>(hB, ei, norm, hC, E);
    epilogue_kernel<64, 6><<<nblk((long long)N * 64), B, 0, stream>>>(
        hB, hC, dinv, b1, g1, be1, rm1, rv1, hA, N);

    // --- layer 2: hA(64) @ W2 -> 64 via WMMA
    {
        long long tiles = (long long)((N + 15) / 16) * (64 / 16);
        gemm_wmma_kernel<64, 64><<<(unsigned)((tiles + 7) / 8), B, 0, stream>>>(hA, W2, hB, N);
    }
    fill_kernel<<<nblk((long long)N * 64), B, 0, stream>>>(hC, (long long)N * 64, 0.0f);
    scatter_kernel<64, 6><<<nblk(E * 64), B, 0, stream>>>(hB, ei, norm, hC, E);
    epilogue_kernel<64, 6><<<nblk((long long)N * 64), B, 0, stream>>>(
        hB, hC, dinv, b2, g2, be2, rm2, rv2, hA, N);

    // --- layer 3: hA(64) @ W3 -> 32 via WMMA
    {
        long long tiles = (long long)((N + 15) / 16) * (32 / 16);
        gemm_wmma_kernel<64, 32><<<(unsigned)((tiles + 7) / 8), B, 0, stream>>>(hA, W3, hB, N);
    }
    fill_kernel<<<nblk((long long)N * 32), B, 0, stream>>>(hC, (long long)N * 32, 0.0f);
    scatter_kernel<32, 5><<<nblk(E * 32), B, 0, stream>>>(hB, ei, norm, hC, E);
    epilogue_kernel<32, 5><<<nblk((long long)N * 32), B, 0, stream>>>(
        hB, hC, dinv, b3, g3, be3, rm3, rv3, hA, N);

    // --- mean pool per graph
    fill_kernel<<<nblk((long long)G * 32), B, 0, stream>>>(pool, (long long)G * 32, 0.0f);
    fill_kernel<<<nblk(G), B, 0, stream>>>(cnt, G, 0.0f);
    pool_kernel<<<nblk((long long)N * 32), B, 0, stream>>>(hA, batch, pool, N);
    cnt_kernel<<<nblk(N), B, 0, stream>>>(batch, cnt, N);
    final_kernel<<<nblk((long long)G * 32), B, 0, stream>>>(pool, cnt, (float*)d_out, G);
}